// ArmaNetwork_24670292149147
// MI455X (gfx1250) — compile-verified
//
#include <hip/hip_runtime.h>
#include <stdint.h>

typedef __attribute__((ext_vector_type(2))) float v2f;
typedef __attribute__((ext_vector_type(8))) float v8f;

#define K_STACKS 2
#define T_ITERS  2
#define F_IN  128
#define HID   64
#define CLS   16

// ---------------------------------------------------------------- utilities

__global__ void zero_f32(float* __restrict__ p, long n) {
  long i = (long)blockIdx.x * blockDim.x + threadIdx.x;
  if (i < n) p[i] = 0.0f;
}

__global__ void degree_kernel(const int* __restrict__ col, unsigned* __restrict__ deg, int E) {
  int e = blockIdx.x * blockDim.x + threadIdx.x;
  if (e < E) atomicAdd(&deg[col[e]], 1u);
}

__global__ void norm_kernel(const int* __restrict__ row, const int* __restrict__ col,
                            const unsigned* __restrict__ deg, float* __restrict__ nrm, int E) {
  int e = blockIdx.x * blockDim.x + threadIdx.x;
  if (e >= E) return;
  float dr = (float)deg[row[e]];
  float dc = (float)deg[col[e]];
  float ir = dr > 0.f ? rsqrtf(dr) : 0.f;
  float ic = dc > 0.f ? rsqrtf(dc) : 0.f;
  nrm[e] = ir * ic;
}

// ------------------------------------------------- fp32 WMMA batched GEMM
// C[bz] (MxN, row-major) = A[bz] (MxK, row-major) @ B[bz] (KxN, row-major)
// One wave per 16x16 C tile, V_WMMA_F32_16X16X4_F32 over K in steps of 4.
// A 16x4 layout : lane m=lane&15 holds K = 2*(lane>>4) + {0,1}  (2 VGPRs/lane)
// B 4x16 layout : lane n=lane&15 holds K = 2*(lane>>4) + {0,1}
// C 16x16 layout: vgpr v, lane l -> row v + 8*(l>>4), col l&15
__global__ void gemm_wmma_f32(const float* __restrict__ A, const float* __restrict__ B,
                              float* __restrict__ C, int Kdim, int Nout, int rowTiles,
                              long aBatch, long bBatch, long cBatch) {
  int wave = threadIdx.x >> 5;
  int lane = threadIdx.x & 31;
  int rowTile = blockIdx.x * (blockDim.x >> 5) + wave;
  if (rowTile >= rowTiles) return;          // uniform per-wave, EXEC stays all-ones
  int colTile = blockIdx.y;
  int bz = blockIdx.z;

  int mn = lane & 15;
  int kb = (lane >> 4) << 1;

  const float* Ab = A + (long)bz * aBatch + (long)(rowTile * 16 + mn) * Kdim;
  const float* Bb = B + (long)bz * bBatch + colTile * 16 + mn;

  v8f acc;
#pragma unroll
  for (int v = 0; v < 8; ++v) acc[v] = 0.0f;

  for (int k0 = 0; k0 < Kdim; k0 += 4) {
    v2f a = *(const v2f*)(Ab + k0 + kb);    // contiguous pair along K
    v2f b;
    b.x = Bb[(long)(k0 + kb) * Nout];
    b.y = Bb[(long)(k0 + kb + 1) * Nout];
    acc = __builtin_amdgcn_wmma_f32_16x16x4_f32(false, a, false, b, (short)0, acc,
                                                false, false);
  }

  float* Cb = C + (long)bz * cBatch + (long)(rowTile * 16) * Nout + colTile * 16 + mn;
  int half = lane >> 4;
#pragma unroll
  for (int v = 0; v < 8; ++v)
    Cb[(long)(v + 8 * half) * Nout] = acc[v];
}

// ---------------------------------------------- sparse propagation (A_hat @ h)
// out[k][col[e]] += h[k][row[e]] * norm[e]; float4 gather, scalar f32 atomics.
__global__ void spmm_scatter(const int* __restrict__ row, const int* __restrict__ col,
                             const float* __restrict__ nrm, const float* __restrict__ H,
                             float* __restrict__ Out, int E, int F, long batch) {
  int k = blockIdx.y;
  long t = (long)blockIdx.x * blockDim.x + threadIdx.x;
  int fch = F >> 2;
  long total = (long)E * fch;
  if (t >= total) return;
  int e  = (int)(t / fch);
  int f4 = (int)(t - (long)e * fch);
  int r = row[e], c = col[e];
  float nv = nrm[e];
  const float4 hv = *(const float4*)(H + (long)k * batch + (long)r * F + f4 * 4);
  float* op = Out + (long)k * batch + (long)c * F + f4 * 4;
  unsafeAtomicAdd(op + 0, hv.x * nv);
  unsafeAtomicAdd(op + 1, hv.y * nv);
  unsafeAtomicAdd(op + 2, hv.z * nv);
  unsafeAtomicAdd(op + 3, hv.w * nv);
}

// out = relu(out + root + bias[k][f]) in place; F is a power of two (64 or 16)
__global__ void relu_add_bias(float* __restrict__ Out, const float* __restrict__ Root,
                              const float* __restrict__ Bias, long NF, int F) {
  int k = blockIdx.y;
  long i = (long)blockIdx.x * blockDim.x + threadIdx.x;
  if (i >= NF) return;
  long idx = (long)k * NF + i;
  int f = (int)(i & (long)(F - 1));
  float v = Out[idx] + Root[idx] + Bias[k * F + f];
  Out[idx] = fmaxf(v, 0.0f);
}

// h = relu(mean over K=2 stacks)
__global__ void mean_relu(const float* __restrict__ In, float* __restrict__ Out, long NF) {
  long i = (long)blockIdx.x * blockDim.x + threadIdx.x;
  if (i >= NF) return;
  float v = 0.5f * (In[i] + In[i + NF]);
  Out[i] = fmaxf(v, 0.0f);
}

__global__ void log_softmax16(const float* __restrict__ In, float* __restrict__ Out, int N) {
  int n = blockIdx.x * blockDim.x + threadIdx.x;
  if (n >= N) return;
  float v[CLS];
  float m = -3.402823466e38f;
#pragma unroll
  for (int i = 0; i < CLS; ++i) { v[i] = In[n * CLS + i]; m = fmaxf(m, v[i]); }
  float s = 0.f;
#pragma unroll
  for (int i = 0; i < CLS; ++i) s += __expf(v[i] - m);
  float ls = __logf(s);
#pragma unroll
  for (int i = 0; i < CLS; ++i) Out[n * CLS + i] = v[i] - m - ls;
}

// ---------------------------------------------------------------- host side

static void run_arma(const float* X, int Fin, int Fout,
                     const float* initW, const float* W, const float* rootW, const float* bias,
                     const int* rowIdx, const int* colIdx, const float* nrm,
                     int N, int E, float* root, float* bufA, float* bufB, float* hOut,
                     hipStream_t stream) {
  int rowTiles = N / 16;                       // N = 50000 -> 3125 tiles exactly
  dim3 gemmGrid((rowTiles + 3) / 4, Fout / 16, K_STACKS);
  dim3 gemmBlock(128);                         // 4 waves, 1 tile per wave
  long NF  = (long)N * Fout;
  long KNF = (long)K_STACKS * NF;

  // root = X @ rootW ; out0 = X @ initW   (A shared across stacks)
  gemm_wmma_f32<<<gemmGrid, gemmBlock, 0, stream>>>(X, rootW, root, Fin, Fout, rowTiles,
                                                    0L, (long)Fin * Fout, NF);
  gemm_wmma_f32<<<gemmGrid, gemmBlock, 0, stream>>>(X, initW, bufA, Fin, Fout, rowTiles,
                                                    0L, (long)Fin * Fout, NF);

  dim3 spGrid((unsigned)(((long)E * (Fout >> 2) + 255) / 256), K_STACKS);
  dim3 rGrid((unsigned)((NF + 255) / 256), K_STACKS);
  for (int t = 0; t < T_ITERS; ++t) {
    if (t > 0) {
      // out = out @ w   (batched over stacks)
      gemm_wmma_f32<<<gemmGrid, gemmBlock, 0, stream>>>(bufB, W, bufA, Fout, Fout, rowTiles,
                                                        NF, (long)Fout * Fout, NF);
    }
    zero_f32<<<(unsigned)((KNF + 255) / 256), 256, 0, stream>>>(bufB, KNF);
    spmm_scatter<<<spGrid, 256, 0, stream>>>(rowIdx, colIdx, nrm, bufA, bufB, E, Fout, NF);
    relu_add_bias<<<rGrid, 256, 0, stream>>>(bufB, root, bias, NF, Fout);
  }
  mean_relu<<<(unsigned)((NF + 255) / 256), 256, 0, stream>>>(bufB, hOut, NF);
}

extern "C" void kernel_launch(void* const* d_in, const int* in_sizes, int n_in,
                              void* d_out, int out_size, void* d_ws, size_t ws_size,
                              hipStream_t stream) {
  (void)n_in; (void)out_size; (void)ws_size;
  const float* x  = (const float*)d_in[0];
  const int*   ei = (const int*)d_in[1];
  const float* p_init[3] = {(const float*)d_in[2],  (const float*)d_in[6],  (const float*)d_in[10]};
  const float* p_w[3]    = {(const float*)d_in[3],  (const float*)d_in[7],  (const float*)d_in[11]};
  const float* p_root[3] = {(const float*)d_in[4],  (const float*)d_in[8],  (const float*)d_in[12]};
  const float* p_b[3]    = {(const float*)d_in[5],  (const float*)d_in[9],  (const float*)d_in[13]};

  int N = in_sizes[0] / F_IN;
  int E = in_sizes[1] / 2;
  const int* rowIdx = ei;
  const int* colIdx = ei + E;

  // -------- workspace carve-out (256B aligned slabs) --------
  uintptr_t base = (uintptr_t)d_ws;
  auto take = [&](size_t bytes) -> void* {
    uintptr_t p = (base + 255) & ~(uintptr_t)255;
    base = p + bytes;
    return (void*)p;
  };
  size_t bufElems = (size_t)K_STACKS * N * HID;
  unsigned* deg  = (unsigned*)take((size_t)N * 4);
  float*    nrm  = (float*)take((size_t)E * 4);
  float*    root = (float*)take(bufElems * 4);
  float*    bufA = (float*)take(bufElems * 4);
  float*    bufB = (float*)take(bufElems * 4);
  float*    h1   = (float*)take((size_t)N * HID * 4);
  float*    h2   = (float*)take((size_t)N * HID * 4);

  // -------- GCN normalization --------
  zero_f32<<<(N + 255) / 256, 256, 0, stream>>>((float*)deg, N);
  degree_kernel<<<(E + 255) / 256, 256, 0, stream>>>(colIdx, deg, E);
  norm_kernel<<<(E + 255) / 256, 256, 0, stream>>>(rowIdx, colIdx, deg, nrm, E);

  // -------- three ARMA conv layers --------
  run_arma(x,  F_IN, HID, p_init[0], p_w[0], p_root[0], p_b[0],
           rowIdx, colIdx, nrm, N, E, root, bufA, bufB, h1, stream);
  run_arma(h1, HID,  HID, p_init[1], p_w[1], p_root[1], p_b[1],
           rowIdx, colIdx, nrm, N, E, root, bufA, bufB, h2, stream);
  run_arma(h2, HID,  CLS, p_init[2], p_w[2], p_root[2], p_b[2],
           rowIdx, colIdx, nrm, N, E, root, bufA, bufB, h1, stream);  // reuse h1 for [N,16]

  // -------- final log_softmax --------
  log_softmax16<<<(N + 255) / 256, 256, 0, stream>>>(h1, (float*)d_out, N);
}